// CCSolver_39831526703347
// MI455X (gfx1250) — compile-verified
//
#include <hip/hip_runtime.h>
#include <hip/hip_bf16.h>
#include <stdint.h>

// Connected-component labeling (8-connectivity) on a 4096x4096 binary image.
// Lock-free union-find (ECL-CC style): init -> merge -> compress -> finalize.
// Root of every tree == min linear index of the component (links always go
// high index -> low index via atomicMin), matching the reference labels
// (min index + 1, background 0) exactly. Labels <= 2^24 are exact in fp32.

#define CCL_H 4096
#define CCL_W 4096
#define NPIX  (CCL_H * CCL_W)

#define TILE_ELEMS 1024   // floats staged per block by the init kernel
#define INIT_BLOCK 256    // 8 wave32 waves

typedef unsigned int u32x4 __attribute__((ext_vector_type(4)));
typedef int          i32x8 __attribute__((ext_vector_type(8)));
typedef int          i32x4 __attribute__((ext_vector_type(4)));

#if defined(__gfx1250__) && \
    __has_builtin(__builtin_amdgcn_tensor_load_to_lds) && \
    __has_builtin(__builtin_amdgcn_s_wait_tensorcnt)
#define CCL_HAVE_TDM 1
#else
#define CCL_HAVE_TDM 0
#endif

// ---------------------------------------------------------------------------
// Kernel 1: threshold prob -> parent array.  parent[i] = i (foreground) or -1.
// On gfx1250 the tile of prob is DMA'd into LDS by the Tensor Data Mover
// (tensor_load_to_lds, TENSORcnt), then consumed as float4 from LDS.
// ---------------------------------------------------------------------------
__global__ void __launch_bounds__(INIT_BLOCK)
ccl_init(const float* __restrict__ prob, int* __restrict__ parent) {
    __shared__ __align__(16) float tile[TILE_ELEMS];
    const int base = (int)blockIdx.x * TILE_ELEMS;

#if CCL_HAVE_TDM
    if (threadIdx.x < 32u) {  // wave 0 issues one TDM DMA for the whole block
        const uint64_t gaddr = (uint64_t)(uintptr_t)(prob + base);
        // Generic LDS addresses carry the LDS byte offset in the low 32 bits.
        const uint32_t laddr = (uint32_t)(uintptr_t)(&tile[0]);

        // D# group 0: count=1 | lds_addr | global_addr[56:0] | type=2 (image)
        const u32x4 g0 = {
            1u,                                     // count=1, user descriptor
            laddr,                                  // lds_addr  (bits 63:32)
            (uint32_t)(gaddr & 0xFFFFFFFFu),        // global_addr[31:0]
            (uint32_t)((gaddr >> 32) & 0x01FFFFFFu) // global_addr[56:32]
                | (2u << 30)                        // D#.type = 2
        };

        // D# group 1: data_size=2 (4B), 1-D tile of TILE_ELEMS elements out of
        // a tensor_dim0 = NPIX "row"; no multicast / pad / iterate / barrier.
        const uint32_t td0 = (uint32_t)NPIX;   // tensor_dim0 (bits 79:48)
        const uint32_t td1 = 1u;               // tensor_dim1 (bits 111:80)
        const i32x8 g1 = {
            (int)(2u << 16),                             // data_size=2 (4 bytes)
            (int)((td0 & 0xFFFFu) << 16),                // tensor_dim0[15:0]
            (int)((td0 >> 16) | ((td1 & 0xFFFFu) << 16)),// td0 hi | td1 lo
            (int)((td1 >> 16) | ((uint32_t)TILE_ELEMS << 16)), // tile_dim0
            0,                                           // tile_dim1=0, tile_dim2=0
            (int)td0,                                    // tensor_dim0_stride[31:0]
            0, 0
        };
        const i32x4 gz4 = {0, 0, 0, 0};            // D# groups 2/3 (unused, <=2D)
        const i32x8 gz8 = {0, 0, 0, 0, 0, 0, 0, 0};

        // 6-arg form (upstream clang-23 / therock-10.0 headers).
        __builtin_amdgcn_tensor_load_to_lds(g0, g1, gz4, gz4, gz8, 0);
        __builtin_amdgcn_s_wait_tensorcnt(0);
    }
    __syncthreads();
    const float4 v = reinterpret_cast<const float4*>(tile)[threadIdx.x];
#else
    const float4 v =
        reinterpret_cast<const float4*>(prob + base)[threadIdx.x];
#endif

    const int i = base + (int)threadIdx.x * 4;
    int4 p;
    p.x = (v.x > 0.5f) ? (i + 0) : -1;
    p.y = (v.y > 0.5f) ? (i + 1) : -1;
    p.z = (v.z > 0.5f) ? (i + 2) : -1;
    p.w = (v.w > 0.5f) ? (i + 3) : -1;
    reinterpret_cast<int4*>(parent)[base / 4 + (int)threadIdx.x] = p;
}

// ---------------------------------------------------------------------------
// Union-find primitives (lock-free; links always high index -> low index,
// so the final root of each component is its minimum linear index).
// ---------------------------------------------------------------------------
__device__ __forceinline__ int ccl_find(int* P, int x) {
    int p = P[x];
    while (p != x) { x = p; p = P[x]; }
    return x;
}

__device__ __forceinline__ void ccl_union(int* P, int a, int b) {
    a = ccl_find(P, a);
    b = ccl_find(P, b);
    while (a != b) {
        if (a > b) { int t = a; a = b; b = t; }      // ensure a < b
        const int old = atomicMin(&P[b], a);         // hook b -> a
        if (old == b) break;                         // b was a root: linked
        b = old;                                     // retry along b's chain
    }
}

// ---------------------------------------------------------------------------
// Kernel 2: union each foreground pixel with its W, NW, N, NE neighbors
// (covers every 8-connectivity edge exactly once).
// ---------------------------------------------------------------------------
__global__ void __launch_bounds__(256)
ccl_merge(int* parent) {
    const int x = (int)(blockIdx.x * blockDim.x + threadIdx.x);
    const int y = (int)(blockIdx.y * blockDim.y + threadIdx.y);
    const int i = y * CCL_W + x;
    if (parent[i] < 0) return;                       // background

    if (x > 0 && parent[i - 1] >= 0) ccl_union(parent, i, i - 1);
    if (y > 0) {
        const int up = i - CCL_W;
        if (x > 0         && parent[up - 1] >= 0) ccl_union(parent, i, up - 1);
        if (                 parent[up]     >= 0) ccl_union(parent, i, up);
        if (x < CCL_W - 1 && parent[up + 1] >= 0) ccl_union(parent, i, up + 1);
    }
}

// ---------------------------------------------------------------------------
// Kernel 3: full path compression -> parent[i] = root(i) for all foreground.
// ---------------------------------------------------------------------------
__global__ void __launch_bounds__(256)
ccl_compress(int* parent) {
    const int i = (int)(blockIdx.x * blockDim.x + threadIdx.x);
    if (parent[i] < 0) return;
    parent[i] = ccl_find(parent, i);
}

// ---------------------------------------------------------------------------
// Kernel 4: convert in place: int root -> float label (root+1), 0 for bg.
// Each thread touches only its own element (compress already flattened),
// and all accesses go through the int view (bitcast the float for storing).
// ---------------------------------------------------------------------------
__global__ void __launch_bounds__(256)
ccl_finalize(int* parent) {
    const int i = (int)(blockIdx.x * blockDim.x + threadIdx.x);
    const int v = parent[i];
    const float label = (v < 0) ? 0.0f : (float)(v + 1);
    parent[i] = __float_as_int(label);
}

// ---------------------------------------------------------------------------
extern "C" void kernel_launch(void* const* d_in, const int* in_sizes, int n_in,
                              void* d_out, int out_size, void* d_ws, size_t ws_size,
                              hipStream_t stream) {
    (void)in_sizes; (void)n_in; (void)out_size; (void)d_ws; (void)ws_size;
    const float* prob = (const float*)d_in[0];
    int* parent = (int*)d_out;   // 16M ints == exactly the 64MB output buffer

    ccl_init<<<NPIX / TILE_ELEMS, INIT_BLOCK, 0, stream>>>(prob, parent);

    dim3 mblk(32, 8);
    dim3 mgrd(CCL_W / 32, CCL_H / 8);
    ccl_merge<<<mgrd, mblk, 0, stream>>>(parent);

    ccl_compress<<<NPIX / 256, 256, 0, stream>>>(parent);
    ccl_finalize<<<NPIX / 256, 256, 0, stream>>>(parent);
}